// AttentionGraphModel_40355512713611
// MI455X (gfx1250) — compile-verified
//
#include <hip/hip_runtime.h>
#include <math.h>

#define N_NODES 50000
#define N_EDGES 800000
#define HID 64
#define HEADS 4

typedef __attribute__((ext_vector_type(2))) float v2f;
typedef __attribute__((ext_vector_type(8))) float v8f;

// ---------------------------------------------------------------------------
// h[n][c] = leaky_relu( sum_k x[n][k] * w[c][k] + b[c], 0.2 )
// One wave32 computes a 16x16 output tile with V_WMMA_F32_16X16X4_F32.
// A (16x4):  lane = 16*half + m ; VGPR v holds K = half*2 + v
// B (4x16):  lane = 16*half + n ; VGPR v holds K = half*2 + v
// C/D (16x16): VGPR r holds M = r + half*8, N = lane&15
// ---------------------------------------------------------------------------
__global__ __launch_bounds__(256) void linear_leaky_wmma(
    const float* __restrict__ x, const float* __restrict__ w,
    const float* __restrict__ b, float* __restrict__ h,
    int K, int nTiles)
{
    const int wave = blockIdx.x * (blockDim.x >> 5) + (threadIdx.x >> 5);
    if (wave >= nTiles) return;                  // wave-uniform: EXEC all-1 for WMMA
    const int rowTile = wave >> 2;               // HID/16 == 4 column tiles
    const int colTile = wave & 3;

    const int lane = threadIdx.x & 31;
    const int half = lane >> 4;                  // selects K pair {0,1} vs {2,3}
    const int lm   = lane & 15;

    const float* xrow = x + (size_t)(rowTile * 16 + lm) * K + half * 2;
    const float* wrow = w + (size_t)(colTile * 16 + lm) * K + half * 2;

    v8f c = {0.f, 0.f, 0.f, 0.f, 0.f, 0.f, 0.f, 0.f};
    for (int k0 = 0; k0 < K; k0 += 4) {
        v2f a, bb;
        a.x  = xrow[k0 + 0]; a.y  = xrow[k0 + 1];
        bb.x = wrow[k0 + 0]; bb.y = wrow[k0 + 1];
        // D = A(16x4) x B(4x16) + C, full fp32
        c = __builtin_amdgcn_wmma_f32_16x16x4_f32(false, a, false, bb,
                                                  (short)0, c, false, false);
    }

    const float bias = b[colTile * 16 + lm];
#pragma unroll
    for (int r = 0; r < 8; ++r) {
        float v = c[r] + bias;
        v = v > 0.f ? v : 0.2f * v;              // leaky_relu(0.2)
        h[(size_t)(rowTile * 16 + r + half * 8) * HID + colTile * 16 + lm] = v;
    }
}

// ---------------------------------------------------------------------------
// s[n][head] = h[n,:] . (aw[head,:64] + aw[head,64:]) + ab[head]
// (HEADS=4: heads 0,1 are the "src" term, heads 2,3 the "dst" term.)
// ---------------------------------------------------------------------------
__global__ __launch_bounds__(256) void scores_kernel(
    const float* __restrict__ h, const float* __restrict__ aw,
    const float* __restrict__ ab, float* __restrict__ s)
{
    int t = blockIdx.x * blockDim.x + threadIdx.x;    // node*4 + head
    if (t >= N_NODES * HEADS) return;
    const int n  = t >> 2;
    const int hd = t & 3;
    const float* hr = h + (size_t)n * HID;
    const float* a0 = aw + hd * (2 * HID);
    float acc = 0.f;
#pragma unroll 8
    for (int f = 0; f < HID; ++f)
        acc += hr[f] * (a0[f] + a0[HID + f]);
    s[t] = acc + ab[hd];
}

// ---------------------------------------------------------------------------
// rowptr[n] = lower_bound(src, n)  (src is sorted); rowptr[N] = E
// ---------------------------------------------------------------------------
__global__ __launch_bounds__(256) void rowptr_kernel(
    const int* __restrict__ src, int* __restrict__ rowptr)
{
    int n = blockIdx.x * blockDim.x + threadIdx.x;
    if (n > N_NODES) return;
    int lo = 0, hi = N_EDGES;
    while (lo < hi) {
        int mid = (lo + hi) >> 1;
        if (src[mid] < n) lo = mid + 1; else hi = mid;
    }
    rowptr[n] = lo;
}

// ---------------------------------------------------------------------------
// Per-node segment softmax + aggregation. One wave32 per node.
// Lane L owns head hd = L&3 and features f = (L>>2) + 8j, j=0..7, so the
// flat output index is n*256 + (f*4+hd) = n*256 + L + 32j  (coalesced).
// FINAL: mean over heads (shfl_xor 1,2) then relu -> (N,64).
// ---------------------------------------------------------------------------
template <bool FINAL>
__global__ __launch_bounds__(256) void aggregate_kernel(
    const float* __restrict__ h,       // (N,64)
    const float* __restrict__ s,       // (N,4) scores incl. att bias
    const int*   __restrict__ dst,
    const int*   __restrict__ rowptr,
    float* __restrict__ out)           // (N,256) or (N,64)
{
    const int n = blockIdx.x * (blockDim.x >> 5) + (threadIdx.x >> 5);
    if (n >= N_NODES) return;
    const int lane = threadIdx.x & 31;
    const int hd = lane & 3;
    const int fb = lane >> 2;          // 0..7
    const int r0 = rowptr[n], r1 = rowptr[n + 1];

    const float sSelf = s[n * HEADS + hd];

    // pass 1: per-head segment max (edges strided by 8 lane-groups)
    float m = -INFINITY;
    for (int i = r0 + fb; i < r1; i += 8) {
        int d = dst[i];
        float e = (hd < 2) ? sSelf : s[d * HEADS + hd];
        m = fmaxf(m, e);
    }
    m = fmaxf(m, __shfl_xor(m, 4, 32));
    m = fmaxf(m, __shfl_xor(m, 8, 32));
    m = fmaxf(m, __shfl_xor(m, 16, 32));

    // pass 2: per-head sum of exp
    float ssum = 0.f;
    for (int i = r0 + fb; i < r1; i += 8) {
        int d = dst[i];
        float e = (hd < 2) ? sSelf : s[d * HEADS + hd];
        ssum += expf(e - m);
    }
    ssum += __shfl_xor(ssum, 4, 32);
    ssum += __shfl_xor(ssum, 8, 32);
    ssum += __shfl_xor(ssum, 16, 32);
    const float inv = (ssum > 0.f) ? 1.f / ssum : 0.f;

    // pass 3: h2[n][f][hd] = sum_edges alpha * h[dst][f]
    float acc[8] = {0, 0, 0, 0, 0, 0, 0, 0};
    for (int i = r0; i < r1; ++i) {
        int d = dst[i];
        float e = (hd < 2) ? sSelf : s[d * HEADS + hd];
        float alpha = expf(e - m) * inv;
        const float* hr = h + (size_t)d * HID;   // L2-resident gather
#pragma unroll
        for (int j = 0; j < 8; ++j)
            acc[j] += hr[fb + 8 * j] * alpha;
    }

    if (!FINAL) {
        float* o = out + (size_t)n * (HID * HEADS);
#pragma unroll
        for (int j = 0; j < 8; ++j)
            o[lane + 32 * j] = fmaxf(acc[j], 0.f);   // relu, matches (N,F*H) reshape
    } else {
#pragma unroll
        for (int j = 0; j < 8; ++j) {
            float v = acc[j];
            v += __shfl_xor(v, 1, 32);
            v += __shfl_xor(v, 2, 32);
            if (hd == 0)
                out[(size_t)n * HID + fb + 8 * j] = fmaxf(v * 0.25f, 0.f);
        }
    }
}

// ---------------------------------------------------------------------------
extern "C" void kernel_launch(void* const* d_in, const int* in_sizes, int n_in,
                              void* d_out, int out_size, void* d_ws, size_t ws_size,
                              hipStream_t stream)
{
    const float* x     = (const float*)d_in[0];
    const float* lw[3] = {(const float*)d_in[1], (const float*)d_in[5], (const float*)d_in[9]};
    const float* lb[3] = {(const float*)d_in[2], (const float*)d_in[6], (const float*)d_in[10]};
    const float* aw[3] = {(const float*)d_in[3], (const float*)d_in[7], (const float*)d_in[11]};
    const float* ab[3] = {(const float*)d_in[4], (const float*)d_in[8], (const float*)d_in[12]};
    const int* src = (const int*)d_in[13];
    const int* dst = (const int*)d_in[14];

    char* ws = (char*)d_ws;
    size_t off = 0;
    auto alloc = [&](size_t bytes) -> void* {
        void* p = ws + off;
        off = (off + bytes + 255) & ~(size_t)255;
        return p;
    };
    float* hbuf   = (float*)alloc((size_t)N_NODES * HID * sizeof(float));
    float* sbuf   = (float*)alloc((size_t)N_NODES * HEADS * sizeof(float));
    int*   rowptr = (int*)  alloc((size_t)(N_NODES + 1) * sizeof(int));
    float* bufA   = (float*)alloc((size_t)N_NODES * HID * HEADS * sizeof(float));
    float* bufB   = (float*)alloc((size_t)N_NODES * HID * HEADS * sizeof(float));

    // CSR row pointers from sorted src (graph topology is per-call constant)
    rowptr_kernel<<<(N_NODES + 1 + 255) / 256, 256, 0, stream>>>(src, rowptr);

    const int nTiles = (N_NODES / 16) * (HID / 16);   // 3125 * 4, exact
    const int linBlocks = (nTiles + 7) / 8;           // 8 waves / block
    const int aggBlocks = (N_NODES + 7) / 8;

    const float* layerIn[3]  = {x, bufA, bufB};
    const int    layerK[3]   = {128, 256, 256};
    float*       layerOut[3] = {bufA, bufB, (float*)d_out};

    for (int L = 0; L < 3; ++L) {
        linear_leaky_wmma<<<linBlocks, 256, 0, stream>>>(
            layerIn[L], lw[L], lb[L], hbuf, layerK[L], nTiles);
        scores_kernel<<<(N_NODES * HEADS + 255) / 256, 256, 0, stream>>>(
            hbuf, aw[L], ab[L], sbuf);
        if (L < 2)
            aggregate_kernel<false><<<aggBlocks, 256, 0, stream>>>(
                hbuf, sbuf, dst, rowptr, layerOut[L]);
        else
            aggregate_kernel<true><<<aggBlocks, 256, 0, stream>>>(
                hbuf, sbuf, dst, rowptr, layerOut[L]);
    }
}